// Extinction_86560770883603
// MI455X (gfx1250) — compile-verified
//
#include <hip/hip_runtime.h>
#include <math.h>

typedef __attribute__((ext_vector_type(2))) float v2f;
typedef __attribute__((ext_vector_type(8))) float v8f;

#define NCH 30
#define PTS_PER_BLOCK 512
#define THREADS 256

// FILTERS rows encoded as channel bitmasks (bit ch set -> filter==1)
__device__ __constant__ unsigned kFiltMask[6] = {
    0x303FFFFFu,  // g0: ch0-21, 28,29
    0x3F3F0003u,  // g1: ch0,1,16-21,24-29
    0x30003333u,  // g2: ch0,1,4,5,8,9,12,13,28,29
    0x3C3FC003u,  // g3: ch0,1,14-21,26-29
    0x30000013u,  // g4: ch0,1,4,28,29
    0x30000CCFu   // g5: ch0-3,6,7,10,11,28,29
};

__global__ __launch_bounds__(THREADS)
void extinction_kernel(const float* __restrict__ tpl,
                       const float* __restrict__ cons,
                       const float* __restrict__ wgas,
                       const float* __restrict__ kW1, const float* __restrict__ kb1,
                       const float* __restrict__ kW2, const float* __restrict__ kb2,
                       const float* __restrict__ kW3, const float* __restrict__ kb3,
                       const float* __restrict__ kWo, const float* __restrict__ kbo,
                       float* __restrict__ out, int N)
{
    __shared__ __align__(16) float lds_ewf[240];              // E[ch*8+j] = exp(w_gas[j,ch]) * filter
    __shared__ __align__(16) float lds_w[432];                // packed MLP weights (426 used)
    __shared__ __align__(16) float lds_m[PTS_PER_BLOCK * 8];  // m[p][j]

    const int tid = threadIdx.x;

    // ---- stage weights + build E in LDS ----
    if (tid < 72)  lds_w[tid]        = kW1[tid];
    if (tid < 36)  lds_w[72  + tid]  = kb1[tid];
    if (tid < 144) lds_w[108 + tid]  = kW2[tid];
    if (tid < 24)  lds_w[252 + tid]  = kb2[tid];
    if (tid < 96)  lds_w[276 + tid]  = kW3[tid];
    if (tid < 24)  lds_w[372 + tid]  = kb3[tid];
    if (tid < 24)  lds_w[396 + tid]  = kWo[tid];
    if (tid < 6)   lds_w[420 + tid]  = kbo[tid];
    if (tid < 240) {
        int ch = tid >> 3, j = tid & 7;
        float v = expf(wgas[j * NCH + ch]);
        if (j >= 2) v = ((kFiltMask[j - 2] >> ch) & 1u) ? v : 0.0f;
        lds_ewf[tid] = v;
    }
    __syncthreads();

    const long blockBase = (long)blockIdx.x * PTS_PER_BLOCK;

    // ---- phase 1: per-point 6x tiny MLP -> s[g]; m[j] = cons[j] * (j<2 ? 1 : s[j-2]) ----
    const float* W1 = lds_w;        const float* B1 = lds_w + 72;
    const float* W2 = lds_w + 108;  const float* B2 = lds_w + 252;
    const float* W3 = lds_w + 276;  const float* B3 = lds_w + 372;
    const float* Wo = lds_w + 396;  const float* Bo = lds_w + 420;

    #pragma unroll
    for (int pp = 0; pp < PTS_PER_BLOCK / THREADS; ++pp) {
        int p = tid + pp * THREADS;
        long n = blockBase + p;
        if (n < N) {
            float t0 = tpl[3 * n], t1 = tpl[3 * n + 1];
            float s[6];
            #pragma unroll
            for (int g = 0; g < 6; ++g) {
                float h1[6];
                #pragma unroll
                for (int o = 0; o < 6; ++o) {
                    float a = fmaf(W1[g*12 + o*2], t0,
                              fmaf(W1[g*12 + o*2 + 1], t1, B1[g*6 + o]));
                    h1[o] = fmaxf(a, 0.0f);
                }
                float h2[4];
                #pragma unroll
                for (int o = 0; o < 4; ++o) {
                    float a = B2[g*4 + o];
                    #pragma unroll
                    for (int k = 0; k < 6; ++k) a = fmaf(W2[g*24 + o*6 + k], h1[k], a);
                    h2[o] = fmaxf(a, 0.0f);
                }
                float h3[4];
                #pragma unroll
                for (int o = 0; o < 4; ++o) {
                    float a = B3[g*4 + o];
                    #pragma unroll
                    for (int k = 0; k < 4; ++k) a = fmaf(W3[g*16 + o*4 + k], h2[k], a);
                    h3[o] = fmaxf(a, 0.0f);
                }
                float x = Bo[g];
                #pragma unroll
                for (int k = 0; k < 4; ++k) x = fmaf(Wo[g*4 + k], h3[k], x);
                s[g] = 1.0f / (1.0f + expf(-x));
            }
            float4 c0 = ((const float4*)cons)[n * 2];
            float4 c1 = ((const float4*)cons)[n * 2 + 1];
            float4 m0 = make_float4(c0.x, c0.y, c0.z * s[0], c0.w * s[1]);
            float4 m1 = make_float4(c1.x * s[2], c1.y * s[3], c1.z * s[4], c1.w * s[5]);
            ((float4*)(lds_m + p * 8))[0] = m0;
            ((float4*)(lds_m + p * 8))[1] = m1;
        }
    }
    __syncthreads();

    // ---- phase 2: OUT(16x240) = M(16x8) @ E'(8x240) via V_WMMA_F32_16X16X4_F32 ----
    // Loops kept ROLLED (#pragma unroll 1): one 8-VGPR accumulator live at a time,
    // low register pressure -> high occupancy to hide NT-store latency.
    const int wave  = tid >> 5;
    const int lane  = tid & 31;
    const int col   = lane & 15;        // A row / B,D column
    const int hi    = lane >> 4;        // lane half: K rows {2hi,2hi+1} (+4 for 1st WMMA)
    const int jl    = lane & 7;         // j of this B column (tile-invariant: 16t % 8 == 0)
    const int chbit = (lane >> 3) & 1;  // channel parity within tile
    const bool nz1a = (jl == 2*hi),     nz1b = (jl == 2*hi + 1);
    const bool nz2a = (jl == 4 + 2*hi), nz2b = (jl == 5 + 2*hi);

    #pragma unroll 1
    for (int c = 0; c < 4; ++c) {
        int  pb = wave * 64 + c * 16;       // block-local point base (uniform per wave)
        long nb = blockBase + pb;
        if (nb + 16 > N) break;             // scalar branch: EXEC stays all-ones for WMMA

        const v2f* mp = (const v2f*)(lds_m + (pb + col) * 8);
        v2f A1 = mp[hi];                    // m[row][{0,1}] | m[row][{2,3}]
        v2f A2 = mp[2 + hi];                // m[row][{4,5}] | m[row][{6,7}]

        float* q = out + (nb + (long)hi * 8) * 240 + col;
        #pragma unroll 1
        for (int t = 0; t < 15; ++t) {
            float ew = lds_ewf[(2*t + chbit) * 8 + jl];
#if __has_builtin(__builtin_amdgcn_wmma_f32_16x16x4_f32)
            v2f B1v = { nz1a ? ew : 0.0f, nz1b ? ew : 0.0f };
            v2f B2v = { nz2a ? ew : 0.0f, nz2b ? ew : 0.0f };
            v8f d = {};
            d = __builtin_amdgcn_wmma_f32_16x16x4_f32(false, A2, false, B2v,
                                                      (short)0, d, false, false);
            d = __builtin_amdgcn_wmma_f32_16x16x4_f32(false, A1, false, B1v,
                                                      (short)0, d, false, false);
            float* qt = q + 16 * t;
            #pragma unroll
            for (int r = 0; r < 8; ++r)
                __builtin_nontemporal_store(d[r], qt + (long)r * 240);
#else
            // scalar fallback: E' has one nonzero per column -> out = m[row][jl] * ew
            float* qt = q + 16 * t;
            #pragma unroll
            for (int r = 0; r < 8; ++r) {
                float v = lds_m[(pb + r + 8 * hi) * 8 + jl] * ew;
                __builtin_nontemporal_store(v, qt + (long)r * 240);
            }
#endif
        }
    }
}

extern "C" void kernel_launch(void* const* d_in, const int* in_sizes, int n_in,
                              void* d_out, int out_size, void* d_ws, size_t ws_size,
                              hipStream_t stream) {
    const float* tpl  = (const float*)d_in[0];
    const float* cons = (const float*)d_in[1];
    const float* wgas = (const float*)d_in[2];
    const float* W1   = (const float*)d_in[3];
    const float* b1   = (const float*)d_in[4];
    const float* W2   = (const float*)d_in[5];
    const float* b2   = (const float*)d_in[6];
    const float* W3   = (const float*)d_in[7];
    const float* b3   = (const float*)d_in[8];
    const float* Wo   = (const float*)d_in[9];
    const float* bo   = (const float*)d_in[10];
    float* out = (float*)d_out;

    int N = in_sizes[0] / 3;
    int blocks = (N + PTS_PER_BLOCK - 1) / PTS_PER_BLOCK;
    extinction_kernel<<<blocks, THREADS, 0, stream>>>(
        tpl, cons, wgas, W1, b1, W2, b2, W3, b3, Wo, bo, out, N);
}